// gwc_block_83159156785844
// MI455X (gfx1250) — compile-verified
//
#include <hip/hip_runtime.h>

typedef __attribute__((ext_vector_type(2))) float v2f;
typedef __attribute__((ext_vector_type(8))) float v8f;

namespace {
constexpr int kB    = 4;
constexpr int kG    = 16;   // N_HEAD
constexpr int kNC   = 20;   // channels per group = 320/16
constexpr int kH    = 96;
constexpr int kW    = 192;
constexpr int kDmax = 64;
constexpr int kTiles = 50;  // 16x16 tiles covering the 64-wide band of the 192x192 Gram matrix
constexpr int kLP   = 200;  // LDS pitch (floats) for L/R rows: 200%64=8 -> conflict-free frag loads, 16B aligned
constexpr int kOP   = 196;  // LDS pitch (floats) for output staging: 16B aligned, near-conflict-free scatter
constexpr float kScale = 0.22360679774997896f;  // 1/sqrt(20)
}

__global__ __launch_bounds__(256)
void gwc_band_wmma(const float* __restrict__ lf, const float* __restrict__ rf,
                   float* __restrict__ out) {
  __shared__ float sL[kNC * kLP];     // 20 x 192 f32 (padded)
  __shared__ float sR[kNC * kLP];
  __shared__ float sO[kDmax * kOP];   // 64 x 192 f32 staging (padded)

  const int h   = blockIdx.x;
  const int g   = blockIdx.y;
  const int b   = blockIdx.z;
  const int tid = threadIdx.x;

  const size_t chStride = (size_t)kH * kW;  // stride between channel planes (floats)
  const size_t inBase   = (((size_t)b * (kG * kNC) + (size_t)g * kNC) * kH + h) * (size_t)kW;

  // ---- stage L and R rows into LDS (float4, fully coalesced) ----
  for (int i = tid; i < kNC * (kW / 4); i += 256) {
    const int c  = i / (kW / 4);
    const int x4 = i % (kW / 4);
    const float4 lv = reinterpret_cast<const float4*>(lf + inBase + c * chStride)[x4];
    const float4 rv = reinterpret_cast<const float4*>(rf + inBase + c * chStride)[x4];
    reinterpret_cast<float4*>(sL + c * kLP)[x4] = lv;
    reinterpret_cast<float4*>(sR + c * kLP)[x4] = rv;
  }
  // ---- zero the band staging tile (covers the x<d zero region of the output) ----
  for (int i = tid; i < kDmax * kOP; i += 256) sO[i] = 0.0f;
  __syncthreads();

  const int wave  = tid >> 5;
  const int lane  = tid & 31;
  const int lmod  = lane & 15;
  const int lhalf = lane >> 4;

  // ---- band tiles of M = L^T R, distributed across the 8 waves ----
  for (int t = wave; t < kTiles; t += 8) {
    int Xi, Yi;
    if (t < 10) {               // triangular corner: Xi in 0..3, Yi in 0..Xi
      int xi = 0;
      while ((xi + 1) * (xi + 2) / 2 <= t) ++xi;
      Xi = xi;
      Yi = t - xi * (xi + 1) / 2;
    } else {                    // full band rows: 5 tiles per Xi, Yi in Xi-4..Xi
      Xi = 4 + (t - 10) / 5;
      Yi = Xi - 4 + (t - 10) % 5;
    }
    const int xA = (Xi << 4) + lmod;   // A-fragment row  (x coordinate)
    const int yB = (Yi << 4) + lmod;   // B-fragment col  (y coordinate)

    v8f acc = {};
#pragma unroll
    for (int kk = 0; kk < 5; ++kk) {   // K = 20 = 5 * 4
      const int c0 = kk * 4 + (lhalf << 1);  // K = 2*(lane/16) + vgpr, per f32 16x4 layout
      v2f av, bv;
      av.x = sL[(c0 + 0) * kLP + xA];
      av.y = sL[(c0 + 1) * kLP + xA];
      bv.x = sR[(c0 + 0) * kLP + yB];
      bv.y = sR[(c0 + 1) * kLP + yB];
      // D = A(16x4,f32) * B(4x16,f32) + C ; full fp32 precision
      acc = __builtin_amdgcn_wmma_f32_16x16x4_f32(false, av, false, bv,
                                                  (short)0, acc, false, false);
    }

    // ---- scatter D[m][n] = M[x][y] into the disparity band: d = x - y ----
#pragma unroll
    for (int r = 0; r < 8; ++r) {
      const int m = r + (lhalf << 3);          // D layout: M = r + 8*(lane/16)
      const int x = (Xi << 4) + m;
      const int d = x - ((Yi << 4) + lmod);    // N = lane%16
      if ((unsigned)d < (unsigned)kDmax) sO[d * kOP + x] = acc[r];
    }
  }
  __syncthreads();

  // ---- coalesced float4 writeout with 1/sqrt(nc) scale ----
  float* op = out + ((((size_t)b * kG + g) * kDmax) * kH + h) * (size_t)kW;
  for (int i = tid; i < kDmax * (kW / 4); i += 256) {
    const int d  = i / (kW / 4);
    const int x4 = i % (kW / 4);
    const float4 v = reinterpret_cast<const float4*>(sO + d * kOP)[x4];
    float4 o;
    o.x = v.x * kScale;
    o.y = v.y * kScale;
    o.z = v.z * kScale;
    o.w = v.w * kScale;
    reinterpret_cast<float4*>(op + d * chStride)[x4] = o;
  }
}

extern "C" void kernel_launch(void* const* d_in, const int* in_sizes, int n_in,
                              void* d_out, int out_size, void* d_ws, size_t ws_size,
                              hipStream_t stream) {
  (void)in_sizes; (void)n_in; (void)d_ws; (void)ws_size; (void)out_size;
  const float* lf = (const float*)d_in[0];
  const float* rf = (const float*)d_in[1];
  float* out = (float*)d_out;
  dim3 grid(kH, kG, kB);   // 96 x 16 x 4 = 6144 workgroups, one per (b, g, h)
  gwc_band_wmma<<<grid, dim3(256), 0, stream>>>(lf, rf, out);
}